// TemporalSelfAttention_48490180772615
// MI455X (gfx1250) — compile-verified
//
#include <hip/hip_runtime.h>
#include <math.h>

typedef __attribute__((ext_vector_type(2))) float v2f;
typedef __attribute__((ext_vector_type(8))) float v8f;

#define DIM_ 64
#define HEADS_ 4
#define HD_ 16
#define B_ 2
#define T_ 288
#define N_ 207
#define QT_ 18          // T/16 row tiles
#define C3_ 192         // 3*DIM

#define KS_STRIDE 20    // padded row stride for K/V LDS tiles (bank-conflict free)
#define S_STRIDE 292    // padded row stride for per-wave S buffer

static __device__ __forceinline__ v8f wmma_f32(v2f a, v2f b, v8f c) {
  // D = A(16x4,f32) * B(4x16,f32) + C(16x16,f32)
  return __builtin_amdgcn_wmma_f32_16x16x4_f32(
      /*neg_a=*/false, a, /*neg_b=*/false, b,
      /*c_mod=*/(short)0, c, /*reuse_a=*/false, /*reuse_b=*/false);
}

// ---------------------------------------------------------------------------
// Kernel A: qkv = Xn @ W_qkv + b_qkv    (rows = B*N*T over t-major tiles)
// q/k/v workspace layout: [b][n][t][64]
// ---------------------------------------------------------------------------
__global__ __launch_bounds__(256) void qkv_proj(
    const float* __restrict__ X, const float* __restrict__ Wqkv,
    const float* __restrict__ bqkv,
    float* __restrict__ q, float* __restrict__ k, float* __restrict__ v) {
  const int lane = threadIdx.x & 31;
  const int wave = threadIdx.x >> 5;
  const int tile = blockIdx.x * 8 + wave;       // 89424 tiles total
  const int rowTile = tile / 12;
  const int colTile = tile % 12;
  const int b   = rowTile / (N_ * QT_);
  const int rem = rowTile % (N_ * QT_);
  const int n   = rem / QT_;
  const int t0  = (rem % QT_) * 16;
  const int col0 = colTile * 16;
  const int lm = lane & 15;
  const int o  = (lane < 16) ? 0 : 2;
  const int t  = t0 + lm;
  const float* xrow = X + ((size_t)(b * T_ + t) * N_ + n) * DIM_;

  v8f acc = {};
#pragma unroll
  for (int s = 0; s < 16; ++s) {
    const int k0 = 4 * s;
    v2f a = *(const v2f*)(xrow + k0 + o);
    v2f bb = { Wqkv[(k0 + o) * C3_ + col0 + lm],
               Wqkv[(k0 + o + 1) * C3_ + col0 + lm] };
    acc = wmma_f32(a, bb, acc);
  }
  const float bias = bqkv[col0 + lm];
  const int mh = (lane < 16) ? 0 : 8;
  const int j = col0 + lm;
  float* dst; int jj;
  if (j < DIM_)            { dst = q; jj = j; }
  else if (j < 2 * DIM_)   { dst = k; jj = j - DIM_; }
  else                     { dst = v; jj = j - 2 * DIM_; }
#pragma unroll
  for (int r = 0; r < 8; ++r)
    dst[((size_t)(b * N_ + n) * T_ + (t0 + r + mh)) * DIM_ + jj] = acc[r] + bias;
}

// ---------------------------------------------------------------------------
// Kernel B: per-(b,n,h) attention. 4 waves; each wave owns 16-row Q tiles.
// O overwrites q workspace in place (same rows/cols, Q already in registers).
// ---------------------------------------------------------------------------
__global__ __launch_bounds__(128) void attention(
    float* qo,                                   // q in, O out (aliased!)
    const float* __restrict__ kmat, const float* __restrict__ vmat) {
  extern __shared__ float smem[];
  const int bnh  = blockIdx.x;                   // (b*N + n)*H + h
  const int h    = bnh % HEADS_;
  const int bn   = bnh / HEADS_;
  const int lane = threadIdx.x & 31;
  const int wave = threadIdx.x >> 5;
  const size_t base = (size_t)bn * T_ * DIM_ + h * HD_;

  float* Ks = smem;
  float* Vs = smem + T_ * KS_STRIDE;
  float* Sw = smem + 2 * T_ * KS_STRIDE + wave * 16 * S_STRIDE;

  // Stage K,V head slices into LDS
  for (int e = threadIdx.x; e < T_ * HD_; e += 128) {
    const int t = e >> 4, d = e & 15;
    Ks[t * KS_STRIDE + d] = kmat[base + (size_t)t * DIM_ + d];
    Vs[t * KS_STRIDE + d] = vmat[base + (size_t)t * DIM_ + d];
  }
  __syncthreads();

  const int lm = lane & 15;
  const int o  = (lane < 16) ? 0 : 2;
  const int mh = (lane < 16) ? 0 : 8;
  const float scale = 0.25f;                     // 1/sqrt(D=16)

  for (int qt = wave; qt < QT_; qt += 4) {
    const int t0 = qt * 16;
    // Q fragments for this 16x16 tile (4 K-steps)
    v2f aQ[4];
    const float* qrow = qo + base + (size_t)(t0 + lm) * DIM_;
#pragma unroll
    for (int s = 0; s < 4; ++s) aQ[s] = *(const v2f*)(qrow + 4 * s + o);

    // S = scale * Q K^T, stored row-major in per-wave LDS buffer
    for (int kt = 0; kt < QT_; ++kt) {
      v8f acc = {};
#pragma unroll
      for (int s = 0; s < 4; ++s) {
        const float* kr = Ks + (kt * 16 + lm) * KS_STRIDE + 4 * s + o;
        v2f bK = { kr[0], kr[1] };
        acc = wmma_f32(aQ[s], bK, acc);
      }
#pragma unroll
      for (int r = 0; r < 8; ++r)
        Sw[(r + mh) * S_STRIDE + kt * 16 + lm] = acc[r] * scale;
    }
    __asm__ volatile("s_wait_dscnt 0" ::: "memory");   // wave-local LDS fence

    // Softmax: lane handles row lm, half (lane>>4) of the 288 columns
    {
      float* row = Sw + lm * S_STRIDE + (lane >> 4) * 144;
      float m = -3.4028235e38f;
      for (int c = 0; c < 144; ++c) m = fmaxf(m, row[c]);
      m = fmaxf(m, __shfl_xor(m, 16, 32));
      float sum = 0.f;
      for (int c = 0; c < 144; ++c) {
        const float p = __expf(row[c] - m);
        row[c] = p;
        sum += p;
      }
      sum += __shfl_xor(sum, 16, 32);
      const float inv = 1.f / sum;
      for (int c = 0; c < 144; ++c) row[c] *= inv;
    }
    __asm__ volatile("s_wait_dscnt 0" ::: "memory");

    // O = P @ V  (contraction over 288 = 18 tiles x 4 steps)
    v8f accO = {};
    for (int st = 0; st < QT_; ++st) {
#pragma unroll
      for (int s = 0; s < 4; ++s) {
        const int kk = st * 16 + 4 * s + o;
        v2f aP = *(const v2f*)(Sw + lm * S_STRIDE + kk);
        v2f bV = { Vs[kk * KS_STRIDE + lm], Vs[(kk + 1) * KS_STRIDE + lm] };
        accO = wmma_f32(aP, bV, accO);
      }
    }
#pragma unroll
    for (int r = 0; r < 8; ++r)
      qo[base + (size_t)(t0 + r + mh) * DIM_ + lm] = accO[r];
  }
}

// ---------------------------------------------------------------------------
// Kernel C: Y[b,t,n,:] = O[b,n,t,:] @ W_out + b_out
// ---------------------------------------------------------------------------
__global__ __launch_bounds__(256) void out_proj(
    const float* __restrict__ O, const float* __restrict__ Wout,
    const float* __restrict__ bout, float* __restrict__ Y) {
  const int lane = threadIdx.x & 31;
  const int wave = threadIdx.x >> 5;
  const int tile = blockIdx.x * 8 + wave;       // 29808 tiles total
  const int rowTile = tile >> 2;
  const int colTile = tile & 3;
  const int b   = rowTile / (N_ * QT_);
  const int rem = rowTile % (N_ * QT_);
  const int n   = rem / QT_;
  const int t0  = (rem % QT_) * 16;
  const int col0 = colTile * 16;
  const int lm = lane & 15;
  const int o  = (lane < 16) ? 0 : 2;
  const float* orow = O + ((size_t)(b * N_ + n) * T_ + (t0 + lm)) * DIM_;

  v8f acc = {};
#pragma unroll
  for (int s = 0; s < 16; ++s) {
    const int k0 = 4 * s;
    v2f a = *(const v2f*)(orow + k0 + o);
    v2f bb = { Wout[(k0 + o) * DIM_ + col0 + lm],
               Wout[(k0 + o + 1) * DIM_ + col0 + lm] };
    acc = wmma_f32(a, bb, acc);
  }
  const float bias = bout[col0 + lm];
  const int mh = (lane < 16) ? 0 : 8;
#pragma unroll
  for (int r = 0; r < 8; ++r) {
    const int t = t0 + r + mh;
    Y[((size_t)(b * T_ + t) * N_ + n) * DIM_ + col0 + lm] = acc[r] + bias;
  }
}

extern "C" void kernel_launch(void* const* d_in, const int* in_sizes, int n_in,
                              void* d_out, int out_size, void* d_ws, size_t ws_size,
                              hipStream_t stream) {
  const float* X    = (const float*)d_in[0];
  const float* Wqkv = (const float*)d_in[1];
  const float* bqkv = (const float*)d_in[2];
  const float* Wout = (const float*)d_in[3];
  const float* bout = (const float*)d_in[4];
  float* Y = (float*)d_out;

  const size_t per = (size_t)B_ * N_ * T_ * DIM_;   // 7,630,848 floats
  float* q = (float*)d_ws;
  float* k = q + per;
  float* v = k + per;

  // A: QKV projection — 89424 tiles / 8 waves per block
  qkv_proj<<<11178, 256, 0, stream>>>(X, Wqkv, bqkv, q, k, v);

  // B: attention — one block per (b,n,h); O overwrites q in place
  const size_t lds = (size_t)(2 * T_ * KS_STRIDE + 4 * 16 * S_STRIDE) * sizeof(float);
  attention<<<B_ * N_ * HEADS_, 128, lds, stream>>>(q, k, v);

  // C: output projection + (B,T,N,C) transpose — 29808 tiles / 8 per block
  out_proj<<<3726, 256, 0, stream>>>(q, Wout, bout, Y);
}